// GATPolicy_7713761264035
// MI455X (gfx1250) — compile-verified
//
#include <hip/hip_runtime.h>
#include <math.h>

// ---------------------------------------------------------------------------
// 2-layer GAT + query projection for MI455X (gfx1250, wave32).
// Edge phases: L2-resident gather/scatter with native fp32/u32 atomics;
//              streamed edge/query index lists use non-temporal loads.
// Matrix phases: V_WMMA_F32_16X16X4_F32 (fp32 WMMA, 16x16 tiles, chained K=4).
// ---------------------------------------------------------------------------

typedef __attribute__((ext_vector_type(2))) float v2f;
typedef __attribute__((ext_vector_type(8))) float v8f;

#define NEG_SLOPE 0.2f
#define EPSV 1e-16f

__device__ __forceinline__ float lrelu(float v) { return v > 0.f ? v : NEG_SLOPE * v; }

// monotonic float<->uint mapping so float max becomes native global_atomic_max_u32
__device__ __forceinline__ unsigned f2ord(float f) {
    unsigned u = __float_as_uint(f);
    return (u & 0x80000000u) ? ~u : (u | 0x80000000u);
}
__device__ __forceinline__ float ord2f(unsigned u) {
    return (u & 0x80000000u) ? __uint_as_float(u & 0x7fffffffu) : __uint_as_float(~u);
}

// --------------------------- layer 1 node init -----------------------------
// h = x * W1 (Fin=1), a_s/a_d, m := leaky(self-loop e), s := 0, acc := 0
__global__ void k_node_init1(const float* __restrict__ x, const float* __restrict__ W1,
                             const float* __restrict__ asw, const float* __restrict__ adw,
                             float* __restrict__ h, float* __restrict__ asn, float* __restrict__ adn,
                             unsigned* __restrict__ mbuf, float* __restrict__ sbuf,
                             float* __restrict__ acc, int N) {
    int n = blockIdx.x * blockDim.x + threadIdx.x;
    if (n >= N) return;
    float xv = x[n];
    float hv[8];
    float as0 = 0.f, as1 = 0.f, ad0 = 0.f, ad1 = 0.f;
#pragma unroll
    for (int j = 0; j < 8; ++j) { hv[j] = xv * W1[j]; h[n * 8 + j] = hv[j]; }
#pragma unroll
    for (int f = 0; f < 4; ++f) {
        as0 += hv[f] * asw[f];     as1 += hv[4 + f] * asw[4 + f];
        ad0 += hv[f] * adw[f];     ad1 += hv[4 + f] * adw[4 + f];
    }
    asn[n * 2] = as0; asn[n * 2 + 1] = as1;
    adn[n * 2] = ad0; adn[n * 2 + 1] = ad1;
    mbuf[n * 2]     = f2ord(lrelu(as0 + ad0));
    mbuf[n * 2 + 1] = f2ord(lrelu(as1 + ad1));
    sbuf[n * 2] = 0.f; sbuf[n * 2 + 1] = 0.f;
#pragma unroll
    for (int j = 0; j < 8; ++j) acc[n * 8 + j] = 0.f;
}

// ------------------ layer 2 node init (h already computed) -----------------
__global__ void k_node_att(const float* __restrict__ h,
                           const float* __restrict__ asw, const float* __restrict__ adw,
                           float* __restrict__ asn, float* __restrict__ adn,
                           unsigned* __restrict__ mbuf, float* __restrict__ sbuf,
                           float* __restrict__ acc, int N) {
    int n = blockIdx.x * blockDim.x + threadIdx.x;
    if (n >= N) return;
    float as0 = 0.f, as1 = 0.f, ad0 = 0.f, ad1 = 0.f;
#pragma unroll
    for (int f = 0; f < 4; ++f) {
        float h0 = h[n * 8 + f], h1 = h[n * 8 + 4 + f];
        as0 += h0 * asw[f];     as1 += h1 * asw[4 + f];
        ad0 += h0 * adw[f];     ad1 += h1 * adw[4 + f];
    }
    asn[n * 2] = as0; asn[n * 2 + 1] = as1;
    adn[n * 2] = ad0; adn[n * 2 + 1] = ad1;
    mbuf[n * 2]     = f2ord(lrelu(as0 + ad0));
    mbuf[n * 2 + 1] = f2ord(lrelu(as1 + ad1));
    sbuf[n * 2] = 0.f; sbuf[n * 2 + 1] = 0.f;
#pragma unroll
    for (int j = 0; j < 8; ++j) acc[n * 8 + j] = 0.f;
}

// ----------------------- edge pass 1: segment max --------------------------
__global__ void k_edge_max(const int* __restrict__ esrc, const int* __restrict__ edst,
                           const float* __restrict__ asn, const float* __restrict__ adn,
                           unsigned* __restrict__ mbuf, int E) {
    int e = blockIdx.x * blockDim.x + threadIdx.x;
    if (e >= E) return;
    int s = __builtin_nontemporal_load(&esrc[e]);   // streamed, no reuse
    int d = __builtin_nontemporal_load(&edst[e]);
    float e0 = lrelu(asn[s * 2]     + adn[d * 2]);
    float e1 = lrelu(asn[s * 2 + 1] + adn[d * 2 + 1]);
    atomicMax(&mbuf[d * 2],     f2ord(e0));
    atomicMax(&mbuf[d * 2 + 1], f2ord(e1));
}

// ------------- edge pass 2: exp, denominator + message scatter -------------
__global__ void k_edge_acc(const int* __restrict__ esrc, const int* __restrict__ edst,
                           const float* __restrict__ asn, const float* __restrict__ adn,
                           const unsigned* __restrict__ mbuf, const float* __restrict__ h,
                           float* __restrict__ sbuf, float* __restrict__ acc, int E) {
    int e = blockIdx.x * blockDim.x + threadIdx.x;
    if (e >= E) return;
    int s = __builtin_nontemporal_load(&esrc[e]);
    int d = __builtin_nontemporal_load(&edst[e]);
    float e0 = lrelu(asn[s * 2]     + adn[d * 2]);
    float e1 = lrelu(asn[s * 2 + 1] + adn[d * 2 + 1]);
    float ex0 = __expf(e0 - ord2f(mbuf[d * 2]));
    float ex1 = __expf(e1 - ord2f(mbuf[d * 2 + 1]));
    unsafeAtomicAdd(&sbuf[d * 2],     ex0);
    unsafeAtomicAdd(&sbuf[d * 2 + 1], ex1);
    const float4 h0 = *(const float4*)(h + (size_t)s * 8);
    const float4 h1 = *(const float4*)(h + (size_t)s * 8 + 4);
    float* a = acc + (size_t)d * 8;
    unsafeAtomicAdd(a + 0, ex0 * h0.x);
    unsafeAtomicAdd(a + 1, ex0 * h0.y);
    unsafeAtomicAdd(a + 2, ex0 * h0.z);
    unsafeAtomicAdd(a + 3, ex0 * h0.w);
    unsafeAtomicAdd(a + 4, ex1 * h1.x);
    unsafeAtomicAdd(a + 5, ex1 * h1.y);
    unsafeAtomicAdd(a + 6, ex1 * h1.z);
    unsafeAtomicAdd(a + 7, ex1 * h1.w);
}

// ------------- node finalize: fold self-loop, divide, add bias -------------
__global__ void k_node_fin(const float* __restrict__ asn, const float* __restrict__ adn,
                           const unsigned* __restrict__ mbuf, const float* __restrict__ sbuf,
                           const float* __restrict__ acc, const float* __restrict__ h,
                           const float* __restrict__ bias, float* __restrict__ out, int N) {
    int n = blockIdx.x * blockDim.x + threadIdx.x;
    if (n >= N) return;
    float m0 = ord2f(mbuf[n * 2]), m1 = ord2f(mbuf[n * 2 + 1]);
    float e0 = lrelu(asn[n * 2]     + adn[n * 2]);
    float e1 = lrelu(asn[n * 2 + 1] + adn[n * 2 + 1]);
    float ex0 = __expf(e0 - m0), ex1 = __expf(e1 - m1);
    float inv0 = 1.f / (sbuf[n * 2] + ex0 + EPSV);
    float inv1 = 1.f / (sbuf[n * 2 + 1] + ex1 + EPSV);
#pragma unroll
    for (int f = 0; f < 4; ++f) {
        out[n * 8 + f]     = (acc[n * 8 + f]     + ex0 * h[n * 8 + f])     * inv0 + bias[f];
        out[n * 8 + 4 + f] = (acc[n * 8 + 4 + f] + ex1 * h[n * 8 + 4 + f]) * inv1 + bias[4 + f];
    }
}

// ---------------- layer-2 linear via WMMA: [16,8] x [8,8] ------------------
// One wave per 16-node tile; K=8 as two chained V_WMMA_F32_16X16X4_F32.
// B operand: unconditional loads from clamped index + v_cndmask (no exec branches).
__global__ void k_linear8x8_wmma(const float* __restrict__ hin, const float* __restrict__ W,
                                 float* __restrict__ hout, int ntiles) {
    int lane = threadIdx.x & 31;
    int wave = threadIdx.x >> 5;
    int tile = blockIdx.x * (blockDim.x >> 5) + wave;
    if (tile >= ntiles) return;                    // wave-uniform, EXEC stays full
    int row = tile * 16 + (lane & 15);
    int kk  = (lane >> 4) * 2;                     // half-wave K split: {0,1} / {2,3}
    v2f a0 = *(const v2f*)(hin + (size_t)row * 8 + kk);
    v2f a1 = *(const v2f*)(hin + (size_t)row * 8 + 4 + kk);
    int nc   = lane & 15;
    bool act = nc < 8;
    int ncc  = act ? nc : 0;                       // clamped column -> always-valid address
    float w00 = W[(kk)     * 8 + ncc], w01 = W[(kk + 1) * 8 + ncc];
    float w10 = W[(4 + kk) * 8 + ncc], w11 = W[(5 + kk) * 8 + ncc];
    v2f b0, b1;
    b0[0] = act ? w00 : 0.f;  b0[1] = act ? w01 : 0.f;
    b1[0] = act ? w10 : 0.f;  b1[1] = act ? w11 : 0.f;
    v8f c = {0.f, 0.f, 0.f, 0.f, 0.f, 0.f, 0.f, 0.f};
    c = __builtin_amdgcn_wmma_f32_16x16x4_f32(false, a0, false, b0, (short)0, c, false, false);
    c = __builtin_amdgcn_wmma_f32_16x16x4_f32(false, a1, false, b1, (short)0, c, false, false);
    if (act) {
        int mbase = tile * 16 + ((lane >> 4) * 8);
#pragma unroll
        for (int v = 0; v < 8; ++v) hout[(size_t)(mbase + v) * 8 + nc] = c[v];
    }
}

__global__ void k_linear_tail(const float* __restrict__ hin, const float* __restrict__ W,
                              float* __restrict__ hout, int start, int N) {
    int n = start + blockIdx.x * blockDim.x + threadIdx.x;
    if (n >= N) return;
#pragma unroll
    for (int j = 0; j < 8; ++j) {
        float acc = 0.f;
#pragma unroll
        for (int k = 0; k < 8; ++k) acc += hin[n * 8 + k] * W[k * 8 + j];
        hout[n * 8 + j] = acc;
    }
}

// -------- final projection via WMMA: concat(h[src],h[dst]) @ Wl[16,2] ------
// One wave per 16-query tile; K=16 as four chained V_WMMA_F32_16X16X4_F32.
__global__ void k_query_wmma(const int* __restrict__ qsrc, const int* __restrict__ qdst,
                             const float* __restrict__ h2, const float* __restrict__ Wl,
                             const float* __restrict__ bl, float* __restrict__ y, int ntiles) {
    int lane = threadIdx.x & 31;
    int wave = threadIdx.x >> 5;
    int tile = blockIdx.x * (blockDim.x >> 5) + wave;
    if (tile >= ntiles) return;
    int row = tile * 16 + (lane & 15);
    int kk  = (lane >> 4) * 2;
    size_t s = (size_t)__builtin_nontemporal_load(&qsrc[row]);  // streamed index lists
    size_t d = (size_t)__builtin_nontemporal_load(&qdst[row]);
    v2f a0 = *(const v2f*)(h2 + s * 8 + kk);       // K 0..3  (src feats 0..3)
    v2f a1 = *(const v2f*)(h2 + s * 8 + 4 + kk);   // K 4..7  (src feats 4..7)
    v2f a2 = *(const v2f*)(h2 + d * 8 + kk);       // K 8..11 (dst feats 0..3)
    v2f a3 = *(const v2f*)(h2 + d * 8 + 4 + kk);   // K 12..15
    int nc   = lane & 15;
    bool act = nc < 2;
    int ncc  = act ? nc : 0;                       // clamped column -> always-valid address
    float w00 = Wl[(kk)      * 2 + ncc], w01 = Wl[(kk + 1)  * 2 + ncc];
    float w10 = Wl[(4 + kk)  * 2 + ncc], w11 = Wl[(5 + kk)  * 2 + ncc];
    float w20 = Wl[(8 + kk)  * 2 + ncc], w21 = Wl[(9 + kk)  * 2 + ncc];
    float w30 = Wl[(12 + kk) * 2 + ncc], w31 = Wl[(13 + kk) * 2 + ncc];
    v2f b0, b1, b2, b3;
    b0[0] = act ? w00 : 0.f;  b0[1] = act ? w01 : 0.f;
    b1[0] = act ? w10 : 0.f;  b1[1] = act ? w11 : 0.f;
    b2[0] = act ? w20 : 0.f;  b2[1] = act ? w21 : 0.f;
    b3[0] = act ? w30 : 0.f;  b3[1] = act ? w31 : 0.f;
    v8f c = {0.f, 0.f, 0.f, 0.f, 0.f, 0.f, 0.f, 0.f};
    c = __builtin_amdgcn_wmma_f32_16x16x4_f32(false, a0, false, b0, (short)0, c, false, false);
    c = __builtin_amdgcn_wmma_f32_16x16x4_f32(false, a1, false, b1, (short)0, c, false, false);
    c = __builtin_amdgcn_wmma_f32_16x16x4_f32(false, a2, false, b2, (short)0, c, false, false);
    c = __builtin_amdgcn_wmma_f32_16x16x4_f32(false, a3, false, b3, (short)0, c, false, false);
    if (act) {
        float bb = bl[nc];
        int mbase = tile * 16 + ((lane >> 4) * 8);
#pragma unroll
        for (int v = 0; v < 8; ++v)
            __builtin_nontemporal_store(c[v] + bb, &y[(size_t)(mbase + v) * 2 + nc]);
    }
}

__global__ void k_query_tail(const int* __restrict__ qsrc, const int* __restrict__ qdst,
                             const float* __restrict__ h2, const float* __restrict__ Wl,
                             const float* __restrict__ bl, float* __restrict__ y,
                             int start, int Q) {
    int q = start + blockIdx.x * blockDim.x + threadIdx.x;
    if (q >= Q) return;
    size_t s = (size_t)qsrc[q], d = (size_t)qdst[q];
    float acc0 = bl[0], acc1 = bl[1];
#pragma unroll
    for (int k = 0; k < 8; ++k) {
        acc0 += h2[s * 8 + k] * Wl[k * 2];
        acc1 += h2[s * 8 + k] * Wl[k * 2 + 1];
        acc0 += h2[d * 8 + k] * Wl[(8 + k) * 2];
        acc1 += h2[d * 8 + k] * Wl[(8 + k) * 2 + 1];
    }
    y[(size_t)q * 2]     = acc0;
    y[(size_t)q * 2 + 1] = acc1;
}

// ---------------------------------------------------------------------------
extern "C" void kernel_launch(void* const* d_in, const int* in_sizes, int n_in,
                              void* d_out, int out_size, void* d_ws, size_t ws_size,
                              hipStream_t stream) {
    const float* x   = (const float*)d_in[0];
    const int*   eix = (const int*)d_in[1];
    const int*   qsrc= (const int*)d_in[2];
    const int*   qdst= (const int*)d_in[3];
    const float* W1  = (const float*)d_in[4];
    const float* as1 = (const float*)d_in[5];
    const float* ad1 = (const float*)d_in[6];
    const float* b1  = (const float*)d_in[7];
    const float* W2  = (const float*)d_in[8];
    const float* as2 = (const float*)d_in[9];
    const float* ad2 = (const float*)d_in[10];
    const float* b2  = (const float*)d_in[11];
    const float* Wl  = (const float*)d_in[12];
    const float* bl  = (const float*)d_in[13];
    float* y = (float*)d_out;

    const int N = in_sizes[0];
    const int E = in_sizes[1] / 2;
    const int Q = in_sizes[2];
    const int* esrc = eix;
    const int* edst = eix + E;

    // workspace: 32*N floats (~25.6 MB @ N=200K) -> fully L2-resident
    float*    ws   = (float*)d_ws;
    float*    hA   = ws;                         // [N,8] current-layer h features
    float*    acc  = ws + (size_t)N * 8;         // [N,8] message accumulator
    float*    hC   = ws + (size_t)N * 16;        // [N,8] layer output
    float*    asn  = ws + (size_t)N * 24;        // [N,2]
    float*    adn  = ws + (size_t)N * 26;        // [N,2]
    unsigned* mbuf = (unsigned*)(ws + (size_t)N * 28); // [N,2] ordered-uint max
    float*    sbuf = ws + (size_t)N * 30;        // [N,2] softmax denominator

    dim3 blk(256);
    int nodeB = (N + 255) / 256;
    int edgeB = (E + 255) / 256;

    // ---------------- layer 1 ----------------
    k_node_init1<<<nodeB, blk, 0, stream>>>(x, W1, as1, ad1, hA, asn, adn, mbuf, sbuf, acc, N);
    k_edge_max <<<edgeB, blk, 0, stream>>>(esrc, edst, asn, adn, mbuf, E);
    k_edge_acc <<<edgeB, blk, 0, stream>>>(esrc, edst, asn, adn, mbuf, hA, sbuf, acc, E);
    k_node_fin <<<nodeB, blk, 0, stream>>>(asn, adn, mbuf, sbuf, acc, hA, b1, hC, N); // hC = h1

    // -------------- layer 2 linear (WMMA) ----
    int lt = N / 16;
    if (lt > 0)
        k_linear8x8_wmma<<<(lt + 7) / 8, blk, 0, stream>>>(hC, W2, hA, lt);          // hA = h2pre
    int lrem = N - lt * 16;
    if (lrem > 0)
        k_linear_tail<<<(lrem + 255) / 256, blk, 0, stream>>>(hC, W2, hA, lt * 16, N);

    // ---------------- layer 2 ----------------
    k_node_att <<<nodeB, blk, 0, stream>>>(hA, as2, ad2, asn, adn, mbuf, sbuf, acc, N);
    k_edge_max <<<edgeB, blk, 0, stream>>>(esrc, edst, asn, adn, mbuf, E);
    k_edge_acc <<<edgeB, blk, 0, stream>>>(esrc, edst, asn, adn, mbuf, hA, sbuf, acc, E);
    k_node_fin <<<nodeB, blk, 0, stream>>>(asn, adn, mbuf, sbuf, acc, hA, b2, hC, N); // hC = h2

    // ---------- final projection (WMMA) ------
    int qt = Q / 16;
    if (qt > 0)
        k_query_wmma<<<(qt + 7) / 8, blk, 0, stream>>>(qsrc, qdst, hC, Wl, bl, y, qt);
    int qrem = Q - qt * 16;
    if (qrem > 0)
        k_query_tail<<<(qrem + 255) / 256, blk, 0, stream>>>(qsrc, qdst, hC, Wl, bl, y, qt * 16, Q);
}